// _InputAttentionRIM_44178033606666
// MI455X (gfx1250) — compile-verified
//
#include <hip/hip_runtime.h>
#include <hip/hip_bf16.h>

typedef float v2f __attribute__((ext_vector_type(2)));
typedef float v8f __attribute__((ext_vector_type(8)));

#define NB     16384
#define NH     1024
#define DKEY   128
#define RDIM   256
#define NK     8
#define LDS_STRIDE (NH + 4)   // pad 4 floats -> conflict-free ds_load_b64 A-fragments

// ---------------------------------------------------------------------------
// Kernel 1: [Kp | V] = x @ [W_k.T | W_v.T]   (16384 x 1024) x (1024 x 384)
// Writes Kp -> workspace, V broadcast x8 -> d_out (V_exp region).
// Block = 256 threads (8 waves), owns 16 rows. Each wave computes 3 col tiles.
// ---------------------------------------------------------------------------
__global__ void __launch_bounds__(256)
kv_kernel(const float* __restrict__ x,
          const float* __restrict__ Wk,   // [128, 1024]
          const float* __restrict__ Wv,   // [256, 1024]
          float* __restrict__ Kp,         // [16384, 128] workspace
          float* __restrict__ vout)       // [16384, 8, 256] V_exp output
{
    extern __shared__ float lds[];                 // 16 x LDS_STRIDE floats
    const int rowBase = blockIdx.x * 16;
    const int tid  = threadIdx.x;

    // ---- stage 16x1024 x-tile into LDS (coalesced float4, one row per iter)
    {
        const int col = tid * 4;                   // 256 thr * 4 floats = 1 row
        #pragma unroll
        for (int i = 0; i < 16; ++i) {
            float4 v = *(const float4*)(x + (size_t)(rowBase + i) * NH + col);
            *(float4*)(lds + i * LDS_STRIDE + col) = v;
        }
    }
    __syncthreads();

    const int wave = tid >> 5;
    const int lane = tid & 31;
    const int nl   = lane & 15;                    // N index within tile / M row
    const int half = lane >> 4;                    // K-half selector

    // A fragment pointer: lane holds row m=nl, K-offset 2*half
    const float* aptr = lds + nl * LDS_STRIDE + 2 * half;

    // B row pointers: columns of the fused [W_k.T | W_v.T] matrix
    const int tbase = wave * 3;                    // 8 waves x 3 tiles = 24 tiles
    const float* bptr[3];
    #pragma unroll
    for (int i = 0; i < 3; ++i) {
        int n = (tbase + i) * 16 + nl;
        const float* p = (n < DKEY) ? (Wk + (size_t)n * NH)
                                    : (Wv + (size_t)(n - DKEY) * NH);
        bptr[i] = p + 2 * half;
    }

    v8f acc[3] = {};
    #pragma unroll 4
    for (int h = 0; h < NH; h += 4) {
        v2f a  = *(const v2f*)(aptr + h);          // ds_load_b64
        #pragma unroll
        for (int i = 0; i < 3; ++i) {
            v2f b = *(const v2f*)(bptr[i] + h);    // global_load_b64 (L2 hit)
            acc[i] = __builtin_amdgcn_wmma_f32_16x16x4_f32(
                         false, a, false, b, (short)0, acc[i], false, false);
        }
    }

    // ---- write results
    #pragma unroll
    for (int i = 0; i < 3; ++i) {
        const int nc = (tbase + i) * 16;
        if (nc < DKEY) {                           // Kp tile -> workspace
            #pragma unroll
            for (int j = 0; j < 8; ++j) {
                int m = j + 8 * half;
                Kp[(size_t)(rowBase + m) * DKEY + nc + nl] = acc[i][j];
            }
        } else {                                   // V tile -> broadcast x8
            const int nv = nc - DKEY + nl;
            #pragma unroll
            for (int j = 0; j < 8; ++j) {
                int m = j + 8 * half;
                float v = acc[i][j];
                size_t base = (size_t)(rowBase + m) * NK * RDIM + nv;
                #pragma unroll
                for (int kk = 0; kk < NK; ++kk)
                    vout[base + (size_t)kk * RDIM] = v;
            }
        }
    }
}

// ---------------------------------------------------------------------------
// Kernel 2: Q[b,k,:] = hidden[b,k,:] @ W_q[k].T (per module), then
// scores[b,k] = (Q[b,k,:] . Kp[b,:]) * D^-0.5.  Q stays in registers.
// Block = 256 threads (8 waves), wave w handles module k=w for 16 rows.
// ---------------------------------------------------------------------------
__global__ void __launch_bounds__(256)
q_scores_kernel(const float* __restrict__ hidden,  // [16384, 8, 256]
                const float* __restrict__ Wq,      // [8, 128, 256]
                const float* __restrict__ Kp,      // [16384, 128]
                float* __restrict__ scores)        // [16384, 8]
{
    const int rowBase = blockIdx.x * 16;
    const int tid  = threadIdx.x;
    const int k    = tid >> 5;                     // module = wave id
    const int lane = tid & 31;
    const int nl   = lane & 15;
    const int half = lane >> 4;

    // A fragment: hidden[rowBase+nl, k, r + 2*half]  (r contiguous -> b64 loads)
    const float* aptr = hidden + ((size_t)(rowBase + nl) * NK + k) * RDIM + 2 * half;
    // B fragment: B[r,d] = W_q[k,d,r]; lane covers d = 16*t + nl
    const float* bbase = Wq + ((size_t)k * DKEY + nl) * RDIM + 2 * half;

    v8f acc[8] = {};                               // 8 tiles cover D=128
    #pragma unroll 2
    for (int r = 0; r < RDIM; r += 4) {
        v2f a = *(const v2f*)(aptr + r);
        #pragma unroll
        for (int t = 0; t < 8; ++t) {
            v2f b = *(const v2f*)(bbase + (size_t)t * 16 * RDIM + r);
            acc[t] = __builtin_amdgcn_wmma_f32_16x16x4_f32(
                         false, a, false, b, (short)0, acc[t], false, false);
        }
    }

    // ---- rowwise dot with Kp: same VGPR layout as the accumulators
    float partial[8];
    #pragma unroll
    for (int j = 0; j < 8; ++j) partial[j] = 0.0f;
    #pragma unroll
    for (int t = 0; t < 8; ++t) {
        #pragma unroll
        for (int j = 0; j < 8; ++j) {
            int m = j + 8 * half;
            float kv = Kp[(size_t)(rowBase + m) * DKEY + 16 * t + nl];
            partial[j] += acc[t][j] * kv;
        }
    }

    const float scale = 0.08838834764831845f;      // 128^-0.5
    #pragma unroll
    for (int j = 0; j < 8; ++j) {
        float p = partial[j];
        p += __shfl_xor(p, 1, 16);                 // reduce across the 16 lanes
        p += __shfl_xor(p, 2, 16);                 //   that share one row
        p += __shfl_xor(p, 4, 16);
        p += __shfl_xor(p, 8, 16);
        if (nl == 0) {
            int m = j + 8 * half;
            scores[(size_t)(rowBase + m) * NK + k] = p * scale;
        }
    }
}

// ---------------------------------------------------------------------------
extern "C" void kernel_launch(void* const* d_in, const int* in_sizes, int n_in,
                              void* d_out, int out_size, void* d_ws, size_t ws_size,
                              hipStream_t stream) {
    const float* x      = (const float*)d_in[0];   // [16384, 1024]
    const float* hidden = (const float*)d_in[1];   // [16384, 8, 256]
    const float* Wq     = (const float*)d_in[2];   // [8, 128, 256]
    const float* Wk     = (const float*)d_in[3];   // [128, 1024]
    const float* Wv     = (const float*)d_in[4];   // [256, 1024]

    float* vexp   = (float*)d_out;                           // [16384,8,256]
    float* scores = (float*)d_out + (size_t)NB * NK * RDIM;  // [16384,8]
    float* Kp     = (float*)d_ws;                            // 8 MB scratch

    const size_t ldsBytes = (size_t)16 * LDS_STRIDE * sizeof(float); // 65792 B

    kv_kernel<<<NB / 16, 256, ldsBytes, stream>>>(x, Wk, Wv, Kp, vexp);
    q_scores_kernel<<<NB / 16, 256, 0, stream>>>(hidden, Wq, Kp, scores);
}